// LNon_20452634263793
// MI455X (gfx1250) — compile-verified
//
#include <hip/hip_runtime.h>
#include <math.h>

// Problem geometry (fixed by the reference): B=16, C=128, H=W=64 -> N = 8,388,608 f32.
#define NTOT   8388608
#define NBLK   2048
#define NTHR   256
// Each thread handles 4 float4 = 16 floats: 2048*256*16 == NTOT exactly (no tail).

typedef float v2f __attribute__((ext_vector_type(2)));
typedef float v8f __attribute__((ext_vector_type(8)));

// ---------------------------------------------------------------------------
// Wave-level (sum, sumsq) reduction via V_WMMA_F32_16X16X4_F32.
// B = all-ones makes every output column identical, so the only layout fact we
// rely on is the f32 C/D layout: lanes 0-15 hold rows 0-7 in
// acc[0..7], lanes 16-31 hold rows 8-15. Sum the 8 accumulator regs per lane
// (= half the wave's total), then one shfl_xor(16) completes the reduction.
// Deterministic fixed-order arithmetic (harness forbids nondeterminism).
// ---------------------------------------------------------------------------
__device__ __forceinline__ void wave_reduce_pair(float s, float q, float& os, float& oq) {
    v2f a1; a1[0] = s; a1[1] = 0.0f;          // 16x4 f32 A operand (2 VGPRs)
    v2f a2; a2[0] = q; a2[1] = 0.0f;
    v2f bb; bb[0] = 1.0f; bb[1] = 1.0f;       // 4x16 ones
    v8f c0 = {};
    v8f d1 = __builtin_amdgcn_wmma_f32_16x16x4_f32(false, a1, false, bb, (short)0, c0, false, false);
    v8f d2 = __builtin_amdgcn_wmma_f32_16x16x4_f32(false, a2, false, bb, (short)0, c0, false, false);
    float rs = 0.0f, rq = 0.0f;
#pragma unroll
    for (int r = 0; r < 8; ++r) { rs += d1[r]; rq += d2[r]; }
    rs += __shfl_xor(rs, 16, 32);
    rq += __shfl_xor(rq, 16, 32);
    os = rs; oq = rq;
}

__device__ __forceinline__ void block_reduce_store(float s, float q,
                                                   float* __restrict__ psum,
                                                   float* __restrict__ psq) {
    float ws_, wq_;
    wave_reduce_pair(s, q, ws_, wq_);         // EXEC all-1s here: no divergence above
    __shared__ float ls[NTHR / 32];
    __shared__ float lq[NTHR / 32];
    const int t = threadIdx.x;
    if ((t & 31) == 0) { ls[t >> 5] = ws_; lq[t >> 5] = wq_; }
    __syncthreads();
    if (t == 0) {
        float bs = 0.0f, bq = 0.0f;
#pragma unroll
        for (int i = 0; i < NTHR / 32; ++i) { bs += ls[i]; bq += lq[i]; }
        psum[blockIdx.x] = bs;
        psq [blockIdx.x] = bq;
    }
}

// ---------------------------------------------------------------------------
// Collapsed pointwise transform.
//   d     = (x - mean) * (iscale/std)                 [ddof=1 std]
//   theta = interp(linspace(-pi,pi,29), 28*sigmoid(d)) == pi*tanh(d/2)
//   ds    = interp(linspace(0,e,29),   28*|tanh(d)|)  == e*|tanh(d)|
//   out   = d*exp(ds*sin(theta)) + ds*cos(theta)
// (softmax row-sums are identically 1 -> integral() == param[idx]).
// ---------------------------------------------------------------------------
__device__ __forceinline__ float pointwise(float x, float mean1, float sc1) {
    const float PI = 3.14159265358979323846f;
    const float EU = 2.71828182845904523536f;
    float d     = (x - mean1) * sc1;
    float theta = PI * tanhf(0.5f * d);
    float dsv   = EU * fabsf(tanhf(d));
    float sn, cs;
    sincosf(theta, &sn, &cs);
    return d * expf(dsv * sn) + dsv * cs;
}

// Pass 1: raw input statistics.
__global__ void __launch_bounds__(NTHR)
k_stats1(const float4* __restrict__ x4, float* __restrict__ psum, float* __restrict__ psq) {
    const size_t base = (size_t)blockIdx.x * (NTHR * 4) + threadIdx.x;
    float s = 0.0f, q = 0.0f;
#pragma unroll
    for (int k = 0; k < 4; ++k) {
        float4 v = x4[base + (size_t)NTHR * k];
        s += v.x + v.y + v.z + v.w;
        q += v.x * v.x + v.y * v.y + v.z * v.z + v.w * v.w;
    }
    block_reduce_store(s, q, psum, psq);
}

// Combine 2048 block partials on one wave (WMMA tree), emit {mean, rstd*scale}.
__global__ void k_final(const float* __restrict__ ps, const float* __restrict__ pq,
                        const float* __restrict__ scale, float* __restrict__ outc) {
    const int L = threadIdx.x;
    float s = 0.0f, q = 0.0f;
#pragma unroll 4
    for (int k = 0; k < NBLK / 32; ++k) { s += ps[L + 32 * k]; q += pq[L + 32 * k]; }
    float ts, tq;
    wave_reduce_pair(s, q, ts, tq);
    if (L == 0) {
        double mean = (double)ts / (double)NTOT;
        double var  = ((double)tq - (double)NTOT * mean * mean) / (double)(NTOT - 1);
        float  rstd = (float)(1.0 / sqrt(var));
        outc[0] = (float)mean;
        outc[1] = rstd * scale[0];
    }
}

// Pass 2: statistics of the (unnormalized) output, recomputed on the fly.
__global__ void __launch_bounds__(NTHR)
k_stats2(const float4* __restrict__ x4, const float* __restrict__ cst,
         float* __restrict__ psum, float* __restrict__ psq) {
    const float m  = cst[0];
    const float sc = cst[1];
    const size_t base = (size_t)blockIdx.x * (NTHR * 4) + threadIdx.x;
    float s = 0.0f, q = 0.0f;
#pragma unroll
    for (int k = 0; k < 4; ++k) {
        float4 v = x4[base + (size_t)NTHR * k];
        float o0 = pointwise(v.x, m, sc);
        float o1 = pointwise(v.y, m, sc);
        float o2 = pointwise(v.z, m, sc);
        float o3 = pointwise(v.w, m, sc);
        s += o0 + o1 + o2 + o3;
        q += o0 * o0 + o1 * o1 + o2 * o2 + o3 * o3;
    }
    block_reduce_store(s, q, psum, psq);
}

// Pass 3: recompute + final normalization, float4 streaming stores.
__global__ void __launch_bounds__(NTHR)
k_write(const float4* __restrict__ x4, const float* __restrict__ cst,
        float4* __restrict__ out4) {
    const float m   = cst[0];
    const float sc  = cst[1];
    const float m2  = cst[2];
    const float sc2 = cst[3];
    const size_t base = (size_t)blockIdx.x * (NTHR * 4) + threadIdx.x;
#pragma unroll
    for (int k = 0; k < 4; ++k) {
        float4 v = x4[base + (size_t)NTHR * k];
        float4 o;
        o.x = (pointwise(v.x, m, sc) - m2) * sc2;
        o.y = (pointwise(v.y, m, sc) - m2) * sc2;
        o.z = (pointwise(v.z, m, sc) - m2) * sc2;
        o.w = (pointwise(v.w, m, sc) - m2) * sc2;
        out4[base + (size_t)NTHR * k] = o;
    }
}

extern "C" void kernel_launch(void* const* d_in, const int* in_sizes, int n_in,
                              void* d_out, int out_size, void* d_ws, size_t ws_size,
                              hipStream_t stream) {
    (void)in_sizes; (void)n_in; (void)out_size; (void)ws_size;
    const float* data   = (const float*)d_in[0];   // (16,128,64,64) f32
    const float* iscale = (const float*)d_in[1];   // scalar
    const float* oscale = (const float*)d_in[2];   // scalar
    // d_in[3..6] (weight_sp, weight_ch, conv_w, conv_b) are dead: softmax
    // row-sums are identically 1, so integral() == param[index].
    float* out = (float*)d_out;

    // Workspace layout (floats): [0..15] constants, then 4x NBLK partials.
    float* wsf = (float*)d_ws;
    float* cst = wsf;                 // [0]=mean1 [1]=iscale/std1 [2]=mean2 [3]=oscale/std2
    float* p1s = wsf + 16;
    float* p1q = p1s + NBLK;
    float* p2s = p1q + NBLK;
    float* p2q = p2s + NBLK;

    k_stats1<<<NBLK, NTHR, 0, stream>>>((const float4*)data, p1s, p1q);
    k_final <<<1, 32, 0, stream>>>(p1s, p1q, iscale, cst + 0);
    k_stats2<<<NBLK, NTHR, 0, stream>>>((const float4*)data, cst, p2s, p2q);
    k_final <<<1, 32, 0, stream>>>(p2s, p2q, oscale, cst + 2);
    k_write <<<NBLK, NTHR, 0, stream>>>((const float4*)data, cst, (float4*)out);
}